// GCNMultiHeadAttentionLayer_1271310320008
// MI455X (gfx1250) — compile-verified
//
#include <hip/hip_runtime.h>
#include <hip/hip_bf16.h>

// Problem constants (match reference)
#define B_ 32
#define N_ 512
#define D_ 256
#define H_ 8
#define HD_ 32
#define E_ 16384
#define NEG_ (-1e10f)

typedef float v2f __attribute__((ext_vector_type(2)));
typedef float v8f __attribute__((ext_vector_type(8)));

__device__ __forceinline__ v8f wmma_f32_k4(v2f a, v2f b, v8f c) {
    // D = A(16x4,f32) * B(4x16,f32) + C(16x16,f32)
    return __builtin_amdgcn_wmma_f32_16x16x4_f32(
        /*neg_a=*/false, a, /*neg_b=*/false, b,
        /*c_mod=*/(short)0, c, /*reuse_a=*/false, /*reuse_b=*/false);
}

// ---------------------------------------------------------------------------
// Degree accumulation over edge destinations (self-loop +1 folded into dinv).
// ---------------------------------------------------------------------------
__global__ __launch_bounds__(256) void deg_kernel(const int* __restrict__ ei,
                                                  float* __restrict__ deg) {
    int e = blockIdx.x * blockDim.x + threadIdx.x;
    if (e < E_) {
        int dst = ei[E_ + e];
        unsafeAtomicAdd(&deg[dst], 1.0f);
    }
}

__global__ __launch_bounds__(256) void dinv_kernel(float* __restrict__ deg) {
    int n = blockIdx.x * blockDim.x + threadIdx.x;
    if (n < N_) deg[n] = rsqrtf(deg[n] + 1.0f);  // +1 self loop
}

// ---------------------------------------------------------------------------
// GEMM: C[M,256] = A[M,256] @ W[256,256] (+ bias).
// Register-blocked: one wave computes a 64x64 output tile as 4x4 WMMA
// 16x16 accumulators (128 acc VGPRs). Per K-step of 4: 4 A-frags + 4 B-frags
// feed 16 V_WMMA_F32_16X16X4_F32 -> ~4x arithmetic intensity vs 16x16/wave.
// ---------------------------------------------------------------------------
__global__ __launch_bounds__(32) void gemm_wmma(const float* __restrict__ A,
                                                const float* __restrict__ W,
                                                const float* __restrict__ bias,
                                                float* __restrict__ C) {
    const int K = D_, NC = D_;
    const int ntn = NC / 64;              // 4 column super-tiles
    int tile = blockIdx.x;
    int tm = tile / ntn;                  // 64-row super-tile
    int tn = tile % ntn;                  // 64-col super-tile
    int lane = threadIdx.x;
    int hf = lane >> 4;                   // half-wave: 0 or 1
    int l  = lane & 15;

    const float* Arow[4];
    const float* Wcol[4];
#pragma unroll
    for (int mi = 0; mi < 4; ++mi)
        Arow[mi] = A + (size_t)(tm * 64 + mi * 16 + l) * K;
#pragma unroll
    for (int ni = 0; ni < 4; ++ni)
        Wcol[ni] = W + (tn * 64 + ni * 16 + l);

    v8f acc[4][4];
#pragma unroll
    for (int mi = 0; mi < 4; ++mi)
#pragma unroll
        for (int ni = 0; ni < 4; ++ni)
            acc[mi][ni] = (v8f){};

    for (int k0 = 0; k0 < K; k0 += 4) {
        int ka = k0 + hf * 2;             // lanes 16-31 hold K+2 (ISA A layout)
        v2f a[4], b[4];
#pragma unroll
        for (int mi = 0; mi < 4; ++mi) {  // contiguous pair -> global_load_b64
            a[mi].x = Arow[mi][ka];
            a[mi].y = Arow[mi][ka + 1];
        }
#pragma unroll
        for (int ni = 0; ni < 4; ++ni) {
            b[ni].x = Wcol[ni][(size_t)ka * NC];
            b[ni].y = Wcol[ni][(size_t)(ka + 1) * NC];
        }
#pragma unroll
        for (int mi = 0; mi < 4; ++mi)
#pragma unroll
            for (int ni = 0; ni < 4; ++ni)
                acc[mi][ni] = wmma_f32_k4(a[mi], b[ni], acc[mi][ni]);
    }

#pragma unroll
    for (int ni = 0; ni < 4; ++ni) {
        int colg = tn * 64 + ni * 16 + l;
        float bv = bias ? bias[colg] : 0.0f;
#pragma unroll
        for (int mi = 0; mi < 4; ++mi) {
#pragma unroll
            for (int j = 0; j < 8; ++j) { // C layout: VGPR j -> row j + hf*8
                int rowg = tm * 64 + mi * 16 + j + hf * 8;
                C[(size_t)rowg * NC + colg] = acc[mi][ni][j] + bv;
            }
        }
    }
}

// ---------------------------------------------------------------------------
// agg = bias + dinv[n]^2 * xw   (self-loop term + bias init)
// ---------------------------------------------------------------------------
__global__ __launch_bounds__(256) void agg_init(const float* __restrict__ xw,
                                                const float* __restrict__ dinv,
                                                const float* __restrict__ bias,
                                                float* __restrict__ agg) {
    int idx = blockIdx.x * blockDim.x + threadIdx.x;
    if (idx >= B_ * N_ * D_) return;
    int d = idx & (D_ - 1);
    int n = (idx >> 8) & (N_ - 1);
    float dn = dinv[n];
    agg[idx] = bias[d] + dn * dn * xw[idx];
}

// ---------------------------------------------------------------------------
// Edge scatter: agg[b,dst,:] += norm(e) * xw[b,src,:].  float4 per thread,
// unsafeAtomicAdd -> global_atomic_add_f32.
// ---------------------------------------------------------------------------
__global__ __launch_bounds__(256) void scatter_edges(const float* __restrict__ xw,
                                                     const int* __restrict__ ei,
                                                     const float* __restrict__ dinv,
                                                     float* __restrict__ agg) {
    size_t idx = (size_t)blockIdx.x * blockDim.x + threadIdx.x;
    if (idx >= (size_t)E_ * B_ * (D_ / 4)) return;
    int d4 = (int)(idx & 63);             // D/4 = 64
    int b  = (int)((idx >> 6) & 31);
    int e  = (int)(idx >> 11);
    int src = ei[e];
    int dst = ei[E_ + e];
    float nrm = dinv[src] * dinv[dst];
    const float4 x = *(const float4*)(xw + ((size_t)b * N_ + src) * D_ + d4 * 4);
    float* o = agg + ((size_t)b * N_ + dst) * D_ + d4 * 4;
    unsafeAtomicAdd(o + 0, nrm * x.x);
    unsafeAtomicAdd(o + 1, nrm * x.y);
    unsafeAtomicAdd(o + 2, nrm * x.z);
    unsafeAtomicAdd(o + 3, nrm * x.w);
}

// ---------------------------------------------------------------------------
// Masked multi-head attention. One wave per (b, h, 16-row tile).
//  energy tiles via WMMA -> mask/scale -> LDS -> softmax -> P@V via WMMA.
// LDS row stride 516 to break bank conflicts in the softmax scans.
// ---------------------------------------------------------------------------
#define LDS_STRIDE 516

__global__ __launch_bounds__(32) void attn_kernel(const float* __restrict__ Q,
                                                  const float* __restrict__ Kt,
                                                  const float* __restrict__ V,
                                                  const int* __restrict__ mask,
                                                  float* __restrict__ out) {
    __shared__ float sE[16 * LDS_STRIDE];   // 33 KB

    int blk = blockIdx.x;                    // b*H*32 + h*32 + rt
    int rt = blk & 31;
    int h  = (blk >> 5) & 7;
    int b  = blk >> 8;
    int lane = threadIdx.x;
    int hf = lane >> 4;
    int l  = lane & 15;
    int r0 = rt * 16;

    const float* Qb = Q + ((size_t)b * N_ + r0) * D_ + h * HD_;
    const float* Kb = Kt + (size_t)b * N_ * D_ + h * HD_;
    const float* Vb = V + (size_t)b * N_ * D_ + h * HD_;

    // Q tile 16x32 held as 8 A-fragments (registers), reused across 32 key tiles
    v2f aQ[8];
#pragma unroll
    for (int ks = 0; ks < 8; ++ks) {
        int kk = ks * 4 + hf * 2;
        aQ[ks].x = Qb[(size_t)l * D_ + kk];
        aQ[ks].y = Qb[(size_t)l * D_ + kk + 1];
    }

    const float inv_scale = 0.17677669529663689f;   // 1/sqrt(HD)

    for (int n0t = 0; n0t < 32; ++n0t) {
        int n0 = n0t * 16;
        const float* Krow = Kb + (size_t)(n0 + l) * D_;
        v8f S = {};
#pragma unroll
        for (int ks = 0; ks < 8; ++ks) {
            int kk = ks * 4 + hf * 2;
            v2f bf; bf.x = Krow[kk]; bf.y = Krow[kk + 1];   // B[kk][n], n = l
            S = wmma_f32_k4(aQ[ks], bf, S);
        }
#pragma unroll
        for (int j = 0; j < 8; ++j) {
            int rloc = j + hf * 8;
            int col  = n0 + l;
            float e = (mask[(size_t)(r0 + rloc) * N_ + col] == 0)
                          ? NEG_ : S[j] * inv_scale;
            sE[rloc * LDS_STRIDE + col] = e;
        }
    }
    __syncthreads();

    // Softmax: lane handles row l, half hf covers columns [hf*256, hf*256+256)
    {
        float* rowp = &sE[l * LDS_STRIDE + hf * 256];
        float mx = -INFINITY;
        for (int c = 0; c < 256; ++c) mx = fmaxf(mx, rowp[c]);
        mx = fmaxf(mx, __shfl_xor(mx, 16, 32));
        float sum = 0.0f;
        for (int c = 0; c < 256; ++c) sum += __expf(rowp[c] - mx);
        sum += __shfl_xor(sum, 16, 32);
        float inv = 1.0f / sum;
        for (int c = 0; c < 256; ++c) rowp[c] = __expf(rowp[c] - mx) * inv;
    }
    __syncthreads();

    // O = P[16,512] @ V[512,32] : A-frags from LDS, two 16-wide N tiles
    v8f acc0 = {}, acc1 = {};
    for (int k0 = 0; k0 < N_; k0 += 4) {
        int kk = k0 + hf * 2;
        v2f ap; ap.x = sE[l * LDS_STRIDE + kk]; ap.y = sE[l * LDS_STRIDE + kk + 1];
        const float* Vr0 = Vb + (size_t)kk * D_;
        const float* Vr1 = Vb + (size_t)(kk + 1) * D_;
        v2f b0; b0.x = Vr0[l];      b0.y = Vr1[l];
        v2f b1; b1.x = Vr0[16 + l]; b1.y = Vr1[16 + l];
        acc0 = wmma_f32_k4(ap, b0, acc0);
        acc1 = wmma_f32_k4(ap, b1, acc1);
    }

    float* Ob = out + ((size_t)b * N_ + r0) * D_ + h * HD_;
#pragma unroll
    for (int j = 0; j < 8; ++j) {
        int rloc = j + hf * 8;
        Ob[(size_t)rloc * D_ + l]      = acc0[j];
        Ob[(size_t)rloc * D_ + 16 + l] = acc1[j];
    }
}

// ---------------------------------------------------------------------------
// Host-side orchestration
// ---------------------------------------------------------------------------
extern "C" void kernel_launch(void* const* d_in, const int* in_sizes, int n_in,
                              void* d_out, int out_size, void* d_ws, size_t ws_size,
                              hipStream_t stream) {
    const float* query = (const float*)d_in[0];
    const float* key_  = (const float*)d_in[1];
    const float* value = (const float*)d_in[2];
    const int*   ei    = (const int*)d_in[3];   // [2,E]
    const int*   mask  = (const int*)d_in[4];   // [N,N]
    const float* Wq = (const float*)d_in[5];  const float* bq = (const float*)d_in[6];
    const float* Wk = (const float*)d_in[7];  const float* bk = (const float*)d_in[8];
    const float* Wv = (const float*)d_in[9];  const float* bv = (const float*)d_in[10];
    const float* Wo = (const float*)d_in[11]; const float* bo = (const float*)d_in[12];
    float* out = (float*)d_out;

    const size_t TSZ = (size_t)B_ * N_ * D_;   // 4,194,304 floats per tensor
    float* ws   = (float*)d_ws;
    float* dinv = ws;                 // 512 (deg -> dinv in place)
    float* xw   = ws + 1024;          // scratch GEMM output / later xctx
    float* Qa   = xw + TSZ;
    float* Ka   = Qa + TSZ;
    float* Va   = Ka + TSZ;

    // degree / norm
    hipMemsetAsync(dinv, 0, N_ * sizeof(float), stream);
    deg_kernel<<<(E_ + 255) / 256, 256, 0, stream>>>(ei, dinv);
    dinv_kernel<<<(N_ + 255) / 256, 256, 0, stream>>>(dinv);

    const int gemm_tiles = (B_ * N_ / 64) * (D_ / 64);            // 1024
    const int elem_blks  = (int)((TSZ + 255) / 256);              // 16384
    const int scat_blks  = (int)(((size_t)E_ * B_ * (D_ / 4) + 255) / 256);

    const float* xs[3]    = { query, key_, value };
    const float* Ws[3]    = { Wq, Wk, Wv };
    const float* bs[3]    = { bq, bk, bv };
    float*       aggs[3]  = { Qa, Ka, Va };

    for (int t = 0; t < 3; ++t) {
        gemm_wmma<<<gemm_tiles, 32, 0, stream>>>(xs[t], Ws[t], nullptr, xw);
        agg_init<<<elem_blks, 256, 0, stream>>>(xw, dinv, bs[t], aggs[t]);
        scatter_edges<<<scat_blks, 256, 0, stream>>>(xw, ei, dinv, aggs[t]);
    }

    // attention -> xctx (reuse xw buffer; V scatter finished before this on stream)
    attn_kernel<<<B_ * H_ * (N_ / 16), 32, 0, stream>>>(Qa, Ka, Va, mask, xw);

    // out = xctx @ Wo + bo
    gemm_wmma<<<gemm_tiles, 32, 0, stream>>>(xw, Wo, bo, out);
}